// TopKSortedAttention_62380105007438
// MI455X (gfx1250) — compile-verified
//
#include <hip/hip_runtime.h>

typedef float v2f __attribute__((ext_vector_type(2)));
typedef float v4f __attribute__((ext_vector_type(4)));
typedef float v8f __attribute__((ext_vector_type(8)));
typedef int   v4i __attribute__((ext_vector_type(4)));

#define DIM       2048
#define NUM_HEADS 128
#define TOPK      32
#define DHV       16
#define BATCH     4
#define SEQ       4096
#define BS        (BATCH * SEQ)   // 16384

#define INFF __builtin_inff()

// ---- CDNA5 async global->LDS path (guarded; falls back to sync staging) ----
#if defined(__gfx1250__) && __has_builtin(__builtin_amdgcn_global_load_async_to_lds_b128)
#define HAVE_ASYNC 1
#else
#define HAVE_ASYNC 0
#endif

#if HAVE_ASYNC
#define GLOBAL_AS __attribute__((address_space(1)))
#define LDS_AS    __attribute__((address_space(3)))
#endif

template <int N>
__device__ __forceinline__ void wait_async() {
#if HAVE_ASYNC
#if __has_builtin(__builtin_amdgcn_s_wait_asynccnt)
    __builtin_amdgcn_s_wait_asynccnt(N);
#else
    asm volatile("s_wait_asynccnt %0" ::"i"(N) : "memory");
#endif
#endif
}

// ---------------------------------------------------------------------------
// GEMM: C[M,N] = A[M,K] * B[N,K]^T   (both row-major, weights stored as rows)
// fp32 WMMA 16x16x4. Block tile 64(M) x 128(N), K staged 32 at a time in LDS,
// double-buffered via GLOBAL_LOAD_ASYNC_TO_LDS_B128 (ASYNCcnt) when available.
// 8 wave32 waves: 2(M) x 4(N), each wave owns a 32x32 C tile (4 accumulators).
// transOut != 0 stores C transposed: C[n*M + m]  (used for q/k -> [h][bs]).
// ---------------------------------------------------------------------------
#define TBM 64
#define TBN 128
#define TBK 32
#define LDSK (TBK + 4)   // 36 floats: 144B row stride (16B aligned, bank-spread)

__device__ __forceinline__ void store_tile(float* __restrict__ C, const v8f& acc,
                                           int M, int N, int row0, int col0,
                                           int lane, int transOut) {
    const int col   = col0 + (lane & 15);
    const int rbase = row0 + ((lane >> 4) << 3);   // lanes 16-31 hold M = r + 8
#pragma unroll
    for (int r = 0; r < 8; ++r) {
        const int row = rbase + r;
        if (transOut) C[(size_t)col * M + row] = acc[r];
        else          C[(size_t)row * N + col] = acc[r];
    }
}

__global__ __launch_bounds__(256, 2)
void gemm_nt_f32(const float* __restrict__ A, const float* __restrict__ Bw,
                 float* __restrict__ C, int M, int N, int K, int transOut)
{
    __shared__ float As[2][TBM * LDSK];
    __shared__ float Bs[2][TBN * LDSK];

    const int tid   = threadIdx.x;
    const int lane  = tid & 31;
    const int wave  = tid >> 5;
    const int waveM = wave >> 2;          // 0..1
    const int waveN = wave & 3;           // 0..3
    const int mh    = lane & 15;          // row within 16
    const int kb2   = (lane >> 4) << 1;   // K sub-offset: 0 or 2 (ISA A/B layout)

    const int m0 = blockIdx.y * TBM;
    const int n0 = blockIdx.x * TBN;

    // stage one K-slab (A: 64x32 = 2 x b128/thread, B: 128x32 = 4 x b128/thread)
    auto stage = [&](int kb, int buf) {
#if HAVE_ASYNC
#pragma unroll
        for (int i = 0; i < 2; ++i) {
            const int idx = tid + i * 256;
            const int r   = idx >> 3;
            const int cv  = (idx & 7) << 2;
            __builtin_amdgcn_global_load_async_to_lds_b128(
                (GLOBAL_AS v4i*)(A + (size_t)(m0 + r) * K + kb + cv),
                (LDS_AS v4i*)(&As[buf][r * LDSK + cv]), 0, 0);
        }
#pragma unroll
        for (int i = 0; i < 4; ++i) {
            const int idx = tid + i * 256;
            const int r   = idx >> 3;
            const int cv  = (idx & 7) << 2;
            __builtin_amdgcn_global_load_async_to_lds_b128(
                (GLOBAL_AS v4i*)(Bw + (size_t)(n0 + r) * K + kb + cv),
                (LDS_AS v4i*)(&Bs[buf][r * LDSK + cv]), 0, 0);
        }
#else
#pragma unroll
        for (int i = 0; i < 2; ++i) {
            const int idx = tid + i * 256;
            const int r   = idx >> 3;
            const int cv  = (idx & 7) << 2;
            v4f d = *(const v4f*)(A + (size_t)(m0 + r) * K + kb + cv);
            *(v4f*)(&As[buf][r * LDSK + cv]) = d;
        }
#pragma unroll
        for (int i = 0; i < 4; ++i) {
            const int idx = tid + i * 256;
            const int r   = idx >> 3;
            const int cv  = (idx & 7) << 2;
            v4f d = *(const v4f*)(Bw + (size_t)(n0 + r) * K + kb + cv);
            *(v4f*)(&Bs[buf][r * LDSK + cv]) = d;
        }
#endif
    };

    v8f acc00 = {}; v8f acc01 = {}; v8f acc10 = {}; v8f acc11 = {};

    stage(0, 0);
    int cur = 0;

    for (int kb = 0; kb < K; kb += TBK) {
        const bool hasNext = (kb + TBK) < K;
        if (hasNext) stage(kb + TBK, cur ^ 1);   // overlaps with compute below

        // wait for the CURRENT stage only: the 6 just-issued ops may remain
        // outstanding (async loads complete in order).
        if (hasNext) wait_async<6>(); else wait_async<0>();
        __syncthreads();

        const float* as = &As[cur][0];
        const float* bs = &Bs[cur][0];
#pragma unroll
        for (int kk = 0; kk < TBK; kk += 4) {
            // A 16x4 frag: lane(0-15)=K{0,1}, lane(16-31)=K{2,3}; B mirrors with N
            v2f a0 = *(const v2f*)(as + (waveM * 32      + mh) * LDSK + kk + kb2);
            v2f a1 = *(const v2f*)(as + (waveM * 32 + 16 + mh) * LDSK + kk + kb2);
            v2f b0 = *(const v2f*)(bs + (waveN * 32      + mh) * LDSK + kk + kb2);
            v2f b1 = *(const v2f*)(bs + (waveN * 32 + 16 + mh) * LDSK + kk + kb2);
            acc00 = __builtin_amdgcn_wmma_f32_16x16x4_f32(false, a0, false, b0, (short)0, acc00, false, false);
            acc01 = __builtin_amdgcn_wmma_f32_16x16x4_f32(false, a0, false, b1, (short)0, acc01, false, false);
            acc10 = __builtin_amdgcn_wmma_f32_16x16x4_f32(false, a1, false, b0, (short)0, acc10, false, false);
            acc11 = __builtin_amdgcn_wmma_f32_16x16x4_f32(false, a1, false, b1, (short)0, acc11, false, false);
        }
        __syncthreads();   // all reads of buf `cur` done before it is re-staged
        cur ^= 1;
    }

    const int rw0 = m0 + waveM * 32;
    const int cl0 = n0 + waveN * 32;
    store_tile(C, acc00, M, N, rw0,      cl0,      lane, transOut);
    store_tile(C, acc01, M, N, rw0,      cl0 + 16, lane, transOut);
    store_tile(C, acc10, M, N, rw0 + 16, cl0,      lane, transOut);
    store_tile(C, acc11, M, N, rw0 + 16, cl0 + 16, lane, transOut);
}

// ---------------------------------------------------------------------------
// Per (b,h): top-32 / bottom-32 keys over S=4096 (permutation within the 32
// doesn't matter: softmax + weighted sum over k is permutation-invariant),
// then gather the 64 selected v rows (16 floats each).
// ---------------------------------------------------------------------------
__global__ __launch_bounds__(256)
void select_topk(const float* __restrict__ k2, const float* __restrict__ vbuf,
                 float* __restrict__ tk, float* __restrict__ bk,
                 float* __restrict__ tv, float* __restrict__ bv)
{
    const int bh  = blockIdx.x;            // b*H + h
    const int b   = bh / NUM_HEADS;
    const int h   = bh % NUM_HEADS;
    const int tid = threadIdx.x;

    __shared__ float keys[SEQ];
    __shared__ float redv[256];
    __shared__ int   redi[256];
    __shared__ int   selIdx[2 * TOPK];

    const float* krow = k2 + (size_t)h * BS + (size_t)b * SEQ;

    for (int i = tid; i < SEQ; i += 256) keys[i] = krow[i];
    __syncthreads();

    // ---- 32 arg-max passes ----
    for (int j = 0; j < TOPK; ++j) {
        float best = -INFF; int bidx = 0;
        for (int i = tid; i < SEQ; i += 256) {
            float v = keys[i];
            if (v > best) { best = v; bidx = i; }
        }
        redv[tid] = best; redi[tid] = bidx;
        __syncthreads();
        for (int off = 128; off > 0; off >>= 1) {
            if (tid < off && redv[tid + off] > redv[tid]) {
                redv[tid] = redv[tid + off]; redi[tid] = redi[tid + off];
            }
            __syncthreads();
        }
        if (tid == 0) {
            selIdx[j] = redi[0];
            tk[(size_t)bh * TOPK + j] = redv[0];
            keys[redi[0]] = -INFF;
        }
        __syncthreads();
    }

    // reload (top entries stay; they can't be bottom-32 since 64 <= 4096)
    for (int i = tid; i < SEQ; i += 256) keys[i] = krow[i];
    __syncthreads();

    // ---- 32 arg-min passes ----
    for (int j = 0; j < TOPK; ++j) {
        float best = INFF; int bidx = 0;
        for (int i = tid; i < SEQ; i += 256) {
            float v = keys[i];
            if (v < best) { best = v; bidx = i; }
        }
        redv[tid] = best; redi[tid] = bidx;
        __syncthreads();
        for (int off = 128; off > 0; off >>= 1) {
            if (tid < off && redv[tid + off] < redv[tid]) {
                redv[tid] = redv[tid + off]; redi[tid] = redi[tid + off];
            }
            __syncthreads();
        }
        if (tid == 0) {
            selIdx[TOPK + j] = redi[0];
            bk[(size_t)bh * TOPK + j] = redv[0];
            keys[redi[0]] = INFF;
        }
        __syncthreads();
    }

    // ---- gather 64 selected v rows: v[b,h,s,:] = vbuf[(b*S+s)*DIM + h*16 + e] ----
    for (int t = tid; t < 2 * TOPK * DHV; t += 256) {
        const int j = t >> 4;
        const int e = t & 15;
        const int s = selIdx[j];
        const float val = vbuf[((size_t)b * SEQ + s) * DIM + h * DHV + e];
        if (j < TOPK) tv[(size_t)bh * TOPK * DHV + j * DHV + e]          = val;
        else          bv[(size_t)bh * TOPK * DHV + (j - TOPK) * DHV + e] = val;
    }
}

// ---------------------------------------------------------------------------
// Rank-1 attention: per (b,h,s) logits = q*key_j, softmax over 32, weighted
// sum of 16-wide rows. q>0 selects branch -> only compute selected branch.
// Writes mid[(b*S+s)*DIM + h*16 + e]  (== transpose(0,2,1,3).reshape layout).
// ---------------------------------------------------------------------------
__global__ __launch_bounds__(256)
void sparse_attn(const float* __restrict__ q2,
                 const float* __restrict__ tk, const float* __restrict__ bk,
                 const float* __restrict__ tv, const float* __restrict__ bv,
                 float* __restrict__ mid)
{
    const int bh  = blockIdx.x;
    const int b   = bh / NUM_HEADS;
    const int h   = bh % NUM_HEADS;
    const int tid = threadIdx.x;

    __shared__ float stk[TOPK], sbk[TOPK];
    __shared__ float stv[TOPK * DHV], sbv[TOPK * DHV];

    if (tid < TOPK) {
        stk[tid] = tk[(size_t)bh * TOPK + tid];
        sbk[tid] = bk[(size_t)bh * TOPK + tid];
    }
    for (int i = tid; i < TOPK * DHV; i += 256) {
        stv[i] = tv[(size_t)bh * TOPK * DHV + i];
        sbv[i] = bv[(size_t)bh * TOPK * DHV + i];
    }
    __syncthreads();

    for (int s = tid; s < SEQ; s += 256) {
        const float qv = q2[(size_t)h * BS + (size_t)b * SEQ + s];
        const float* kk = (qv > 0.0f) ? stk : sbk;
        const float* vv = (qv > 0.0f) ? stv : sbv;

        float mx = -INFF;
#pragma unroll
        for (int j = 0; j < TOPK; ++j) mx = fmaxf(mx, qv * kk[j]);

        float o[DHV];
#pragma unroll
        for (int e = 0; e < DHV; ++e) o[e] = 0.0f;
        float denom = 0.0f;
#pragma unroll
        for (int j = 0; j < TOPK; ++j) {
            const float w = __expf(qv * kk[j] - mx);
            denom += w;
#pragma unroll
            for (int e = 0; e < DHV; ++e) o[e] += w * vv[j * DHV + e];
        }
        const float inv = 1.0f / denom;
        float* dst = mid + ((size_t)b * SEQ + s) * DIM + h * DHV;
#pragma unroll
        for (int e = 0; e < DHV; ++e) dst[e] = o[e] * inv;
    }
}

// ---------------------------------------------------------------------------
extern "C" void kernel_launch(void* const* d_in, const int* in_sizes, int n_in,
                              void* d_out, int out_size, void* d_ws, size_t ws_size,
                              hipStream_t stream)
{
    const float* x  = (const float*)d_in[0];
    const float* Wq = (const float*)d_in[1];
    const float* Wk = (const float*)d_in[2];
    const float* Wv = (const float*)d_in[3];
    const float* Wo = (const float*)d_in[4];
    float* out = (float*)d_out;

    // workspace carve-up (floats). vbuf (128MB) is reused as `mid` after the
    // select kernel has gathered the 64 rows it needs per (b,h).
    float* ws   = (float*)d_ws;
    float* q2   = ws;                                            // H*BS      (8MB)
    float* k2   = q2 + (size_t)NUM_HEADS * BS;                   // H*BS      (8MB)
    float* tk   = k2 + (size_t)NUM_HEADS * BS;                   // B*H*32
    float* bk   = tk + (size_t)BATCH * NUM_HEADS * TOPK;         // B*H*32
    float* tv   = bk + (size_t)BATCH * NUM_HEADS * TOPK;         // B*H*32*16 (1MB)
    float* bv   = tv + (size_t)BATCH * NUM_HEADS * TOPK * DHV;   // B*H*32*16 (1MB)
    float* vbuf = bv + (size_t)BATCH * NUM_HEADS * TOPK * DHV;   // BS*DIM    (128MB)

    const dim3 blk(256);

    // q = x @ Wq^T  -> stored transposed q2[h*BS + bs]
    gemm_nt_f32<<<dim3(NUM_HEADS / TBN, BS / TBM), blk, 0, stream>>>(
        x, Wq, q2, BS, NUM_HEADS, DIM, 1);
    // k = x @ Wk^T  -> k2[h*BS + bs]
    gemm_nt_f32<<<dim3(NUM_HEADS / TBN, BS / TBM), blk, 0, stream>>>(
        x, Wk, k2, BS, NUM_HEADS, DIM, 1);
    // v = x @ Wv^T  -> vbuf[bs*DIM + e]
    gemm_nt_f32<<<dim3(DIM / TBN, BS / TBM), blk, 0, stream>>>(
        x, Wv, vbuf, BS, DIM, DIM, 0);

    // top/bottom-32 selection + v-row gather, per (b,h)
    select_topk<<<dim3(BATCH * NUM_HEADS), blk, 0, stream>>>(k2, vbuf, tk, bk, tv, bv);

    // attention; writes `mid` into vbuf (v no longer needed)
    sparse_attn<<<dim3(BATCH * NUM_HEADS), blk, 0, stream>>>(q2, tk, bk, tv, bv, vbuf);

    // out = mid @ Wo^T
    gemm_nt_f32<<<dim3(DIM / TBN, BS / TBM), blk, 0, stream>>>(
        vbuf, Wo, out, BS, DIM, DIM, 0);
}